// ProviderGAT_72430328480258
// MI455X (gfx1250) — compile-verified
//
#include <hip/hip_runtime.h>

#define CODE_DIM 256
#define HIDDEN   128
#define HEADS    4
#define N_CODES  100000
#define N_SPEC   256

#define TILE_CODES 128
#define TSTR       260        // padded LDS row stride (floats): banks = (4*row + d) % 64
#define NBLOCKS    391        // 782 tiles -> exactly 2 tiles per block
#define THREADS    256
#define NWAVES     8
#define REC_STRIDE 1032       // 4 m + 4 Z + 1024 s per block record

typedef float v2f __attribute__((ext_vector_type(2)));
typedef float v8f __attribute__((ext_vector_type(8)));
typedef unsigned int u32x4 __attribute__((ext_vector_type(4)));
typedef int i32x4 __attribute__((ext_vector_type(4)));
typedef int i32x8 __attribute__((ext_vector_type(8)));

// LDS layout (floats) -- double-buffered tiles for TDM prefetch
#define LDS_TILE0  0
#define LDS_TILE1  (TILE_CODES * TSTR)                 // 33280
#define LDS_V      (2 * TILE_CODES * TSTR)             // 66560
#define LDS_EBUF   (LDS_V + HEADS * TSTR)              // 67600
#define LDS_MBUF   (LDS_EBUF + NWAVES * 16 * HEADS)    // 68112
#define LDS_C      (LDS_MBUF + NWAVES * HEADS)         // 68144
#define LDS_STAT   (LDS_C + HEADS)                     // 68148
#define LDS_TOTAL  (LDS_STAT + NWAVES * 2 * HEADS)     // 68212 floats = 272,848 B (<320KB WGP LDS)

#if __has_builtin(__builtin_amdgcn_tensor_load_to_lds) && __has_builtin(__builtin_amdgcn_s_wait_tensorcnt)
#define HAVE_TDM 1
#else
#define HAVE_TDM 0
#endif

__device__ __forceinline__ v8f wmma_f32_16x16x4(v2f a, v2f b, v8f c) {
  // D = A(16x4) * B(4x16) + C(16x16), fp32, wave32
  return __builtin_amdgcn_wmma_f32_16x16x4_f32(false, a, false, b, (short)0, c, false, false);
}

#if HAVE_TDM
// Issue one TDM load: tile of TILE_CODES x CODE_DIM f32 from global to LDS.
// Hardware applies: zero-fill for rows >= rem_rows (tensor_dim1 OOB), and
// LDS padding of 4 DWORDs per 256 DWORDs -> padded row stride of 260 floats.
__device__ __forceinline__ void tdm_load_tile(unsigned lds_byte_addr,
                                              const float* gptr, int rem_rows) {
  const unsigned long long ga = (unsigned long long)(size_t)gptr;
  u32x4 g0;
  g0.x = 1u;                                             // count=1 (valid user D#)
  g0.y = lds_byte_addr;                                  // lds_addr [63:32]
  g0.z = (unsigned)(ga & 0xFFFFFFFFu);                   // global_addr [95:64]
  g0.w = (unsigned)((ga >> 32) & 0x01FFFFFFu)            // global_addr [120:96]
       | 0x80000000u;                                    // type=2 ("image") [127:126]
  i32x8 g1;
  g1[0] = (int)((2u << 16)      // data_size = 4 bytes
              | (1u << 20)      // pad_enable
              | (7u << 22)      // pad_interval code 7 = every 256 DWORDs
              | (3u << 25));    // pad_amount  code 3 = 4 DWORDs  -> stride 260
  g1[1] = (int)(((unsigned)CODE_DIM & 0xFFFFu) << 16);   // tensor_dim0 lo16 @ bits 63:48
  g1[2] = (int)(((unsigned)rem_rows & 0xFFFFu) << 16);   // dim0 hi16 = 0 | tensor_dim1 lo16
  g1[3] = (int)((((unsigned)rem_rows >> 16) & 0xFFFFu)   // tensor_dim1 hi16
              | ((unsigned)CODE_DIM << 16));             // tile_dim0 = 256
  g1[4] = TILE_CODES;                                    // tile_dim1 = 128 | tile_dim2 = 0
  g1[5] = CODE_DIM;                                      // tensor_dim0_stride = 256 (lo32)
  g1[6] = 0;                                             // stride hi | dim1_stride lo
  g1[7] = 0;
  i32x4 gz4; gz4[0] = 0; gz4[1] = 0; gz4[2] = 0; gz4[3] = 0;   // groups 2/3 unused (2D)
  i32x8 gz8;
#pragma unroll
  for (int i = 0; i < 8; i++) gz8[i] = 0;
  // 6-arg toolchain form: (g0, g1, g2, g3, extra, cpol)
  __builtin_amdgcn_tensor_load_to_lds(g0, g1, gz4, gz4, gz8, 0);
}
#endif

// ---------------------------------------------------------------------------
// Kernel 1: precompute v[h][d] = W[h,d,:] . a2[h]   (4x256)
//           and c[h] = provider_h[h] . a1[h] + b[h] . a2[h]
// ws layout: v[0..1023], c[1024..1027]
// ---------------------------------------------------------------------------
__global__ __launch_bounds__(THREADS) void gat_prep_kernel(
    const float* __restrict__ pe, const float* __restrict__ W,
    const float* __restrict__ b,  const float* __restrict__ a,
    float* __restrict__ ws)
{
  __shared__ float ph_s[HEADS * HIDDEN];
  const int t = threadIdx.x;

  for (int idx = t; idx < HEADS * CODE_DIM; idx += THREADS) {
    const int h = idx >> 8, d = idx & 255;
    const float* wrow = W + ((size_t)h * CODE_DIM + d) * HIDDEN;
    const float* a2   = a + h * 2 * HIDDEN + HIDDEN;
    float s = 0.f;
    for (int k = 0; k < HIDDEN; k++) s += wrow[k] * a2[k];
    ws[idx] = s;                                   // v[h][d]
  }
  for (int idx = t; idx < HEADS * HIDDEN; idx += THREADS) {
    const int h = idx >> 7, k = idx & (HIDDEN - 1);
    float s = b[idx];
    const float* wp = W + (size_t)h * CODE_DIM * HIDDEN + k;
    for (int d = 0; d < CODE_DIM; d++) s += pe[d] * wp[(size_t)d * HIDDEN];
    ph_s[idx] = s;                                 // provider_h[h][k]
  }
  __syncthreads();
  if (t < HEADS) {
    const float* a1 = a + t * 2 * HIDDEN;
    const float* a2 = a1 + HIDDEN;
    float s = 0.f;
    for (int k = 0; k < HIDDEN; k++)
      s += ph_s[t * HIDDEN + k] * a1[k] + b[t * HIDDEN + k] * a2[k];
    ws[HEADS * CODE_DIM + t] = s;                  // c[h]
  }
}

// ---------------------------------------------------------------------------
// Kernel 2: single-pass online-softmax stream over code_embs.
// TDM double-buffered tiles (128 codes x 256 dims, padded stride 260 in LDS);
// wave 0 drives the DMA, all 8 waves compute with WMMA f32 16x16x4.
// Phase E:  e(16 codes x 4 heads)  = tile_slice @ v^T   (64 x wmma)
// Phase W:  s(4 heads x 256 dims) += w @ tile_slice     (64 x wmma)
// Block writes one record (m[4], Z[4], s[4][256]); deterministic combine.
// ---------------------------------------------------------------------------
__global__ __launch_bounds__(THREADS) void gat_stream_kernel(
    const float* __restrict__ codes, const float* __restrict__ vc,
    float* __restrict__ recs)
{
  extern __shared__ float lds[];
  float* v_lds = lds + LDS_V;
  float* e_buf = lds + LDS_EBUF;
  float* m_buf = lds + LDS_MBUF;
  float* c_lds = lds + LDS_C;
  float* stat  = lds + LDS_STAT;

  const int t     = threadIdx.x;
  const int lane  = t & 31;
  const int wv    = t >> 5;
  const int n     = lane & 15;     // WMMA N / M-row index within half
  const int half  = lane >> 4;     // selects K pair {0,1} vs {2,3}
  const int rbase = wv * 16;       // this wave's code rows within a tile

  for (int idx = t; idx < HEADS * CODE_DIM; idx += THREADS)
    v_lds[(idx >> 8) * TSTR + (idx & 255)] = vc[idx];
  if (t < HEADS) c_lds[t] = vc[HEADS * CODE_DIM + t];

  float m_run[HEADS], z_run[HEADS];
#pragma unroll
  for (int h = 0; h < HEADS; h++) { m_run[h] = -3.0e38f; z_run[h] = 0.f; }

  v8f acc_s[16];                   // persistent: heads(rows 0..3) x 256 dims
#pragma unroll
  for (int cg = 0; cg < 16; cg++)
#pragma unroll
    for (int r = 0; r < 8; r++) acc_s[cg][r] = 0.f;

  const int ntiles = (N_CODES + TILE_CODES - 1) / TILE_CODES;
  int bufsel = 0;

#if HAVE_TDM
  const unsigned ldsDynBase = (unsigned)__builtin_amdgcn_groupstaticsize();
  // Prologue: prefetch this block's first tile into buffer 0.
  if (wv == 0) {
    const int base0 = blockIdx.x * TILE_CODES;
    tdm_load_tile(ldsDynBase + LDS_TILE0 * 4u,
                  codes + (size_t)base0 * CODE_DIM, N_CODES - base0);
  }
#endif

  for (int tileIdx = blockIdx.x; tileIdx < ntiles; tileIdx += gridDim.x) {
    const int base = tileIdx * TILE_CODES;
    float* tile = lds + (bufsel ? LDS_TILE1 : LDS_TILE0);

#if HAVE_TDM
    if (wv == 0) __builtin_amdgcn_s_wait_tensorcnt(0);  // current tile landed
    __syncthreads();   // tile visible to all waves; prev compute on other buf done
    const int nextTile = tileIdx + gridDim.x;
    if (nextTile < ntiles && wv == 0) {                 // prefetch into other buf
      const int nbase = nextTile * TILE_CODES;
      tdm_load_tile(ldsDynBase + (bufsel ? LDS_TILE0 : LDS_TILE1) * 4u,
                    codes + (size_t)nbase * CODE_DIM, N_CODES - nbase);
    }
#else
    __syncthreads();   // previous tile fully consumed
#pragma unroll 4
    for (int j = 0; j < 32; j++) {
      const int idx  = j * THREADS + t;
      const int row  = idx >> 6;
      const int col4 = (idx & 63) << 2;
      float4 val = make_float4(0.f, 0.f, 0.f, 0.f);
      const int code = base + row;
      if (code < N_CODES)
        val = *(const float4*)(codes + (size_t)code * CODE_DIM + col4);
      *(float4*)(tile + row * TSTR + col4) = val;
    }
    __syncthreads();
#endif

    // ---- Phase E: scores via WMMA (4 independent accumulators) ----
    v8f accE[4];
#pragma unroll
    for (int u = 0; u < 4; u++)
#pragma unroll
      for (int r = 0; r < 8; r++) accE[u][r] = 0.f;

    const int arow = (rbase + n) * TSTR;
    for (int ks4 = 0; ks4 < 16; ks4++) {
#pragma unroll
      for (int u = 0; u < 4; u++) {
        const int d = (ks4 * 4 + u) * 4 + 2 * half;
        v2f A = *(const v2f*)(tile + arow + d);          // A[M=n][K=2h..2h+1]
        v2f B; B.x = 0.f; B.y = 0.f;
        if (n < HEADS) B = *(const v2f*)(v_lds + n * TSTR + d); // B[K][N=n]
        accE[u] = wmma_f32_16x16x4(A, B, accE[u]);
      }
    }
    v8f acc_e = accE[0] + accE[1] + accE[2] + accE[3];

    // C layout: elem r = e[code rbase + r + 8*half][head n] (n<4 meaningful)
    if (n < HEADS) {
      const float ch = c_lds[n];
#pragma unroll
      for (int r = 0; r < 8; r++) {
        float e = acc_e[r] + ch;
        e = (e >= 0.f) ? e : 0.2f * e;                   // leaky relu
        e_buf[(rbase + r + 8 * half) * HEADS + n] = e;
      }
    }

    // ---- online softmax stats over this wave's 16 codes ----
    float tmax[HEADS];
#pragma unroll
    for (int h = 0; h < HEADS; h++) tmax[h] = -3.0e38f;
    for (int i = 0; i < 16; i++) {
      const bool ok = (base + rbase + i) < N_CODES;
#pragma unroll
      for (int h = 0; h < HEADS; h++) {
        const float e = ok ? e_buf[(rbase + i) * HEADS + h] : -3.0e38f;
        tmax[h] = fmaxf(tmax[h], e);
      }
    }
    float m_new[HEADS], scl[HEADS], sumw[HEADS];
#pragma unroll
    for (int h = 0; h < HEADS; h++) {
      m_new[h] = fmaxf(m_run[h], tmax[h]);
      scl[h]   = __expf(m_run[h] - m_new[h]);
      sumw[h]  = 0.f;
    }
    for (int i = 0; i < 16; i++) {
      const bool ok = (base + rbase + i) < N_CODES;
#pragma unroll
      for (int h = 0; h < HEADS; h++) {
        const float w = ok ? __expf(e_buf[(rbase + i) * HEADS + h] - m_new[h]) : 0.f;
        sumw[h] += w;
      }
    }
#pragma unroll
    for (int h = 0; h < HEADS; h++) {
      z_run[h] = z_run[h] * scl[h] + sumw[h];
      m_run[h] = m_new[h];
    }
    if (lane == 0) {
      m_buf[wv * HEADS + 0] = m_new[0]; m_buf[wv * HEADS + 1] = m_new[1];
      m_buf[wv * HEADS + 2] = m_new[2]; m_buf[wv * HEADS + 3] = m_new[3];
    }
#pragma unroll
    for (int cg = 0; cg < 16; cg++) {                    // rescale accumulator
      acc_s[cg][0] *= scl[0]; acc_s[cg][1] *= scl[1];
      acc_s[cg][2] *= scl[2]; acc_s[cg][3] *= scl[3];
    }

    // ---- Phase W: s += w(4x16 codes) @ tile_slice(16 codes x 256 dims) ----
#pragma unroll
    for (int kst = 0; kst < 4; kst++) {
      const int c0 = kst * 4 + 2 * half;                 // A K-codes for this lane
      v2f Aw; Aw.x = 0.f; Aw.y = 0.f;
      if (n < HEADS) {
        const float mh = m_buf[wv * HEADS + n];
        const bool ok0 = (base + rbase + c0    ) < N_CODES;
        const bool ok1 = (base + rbase + c0 + 1) < N_CODES;
        const float e0 = e_buf[(rbase + c0    ) * HEADS + n];
        const float e1 = e_buf[(rbase + c0 + 1) * HEADS + n];
        Aw.x = ok0 ? __expf(e0 - mh) : 0.f;
        Aw.y = ok1 ? __expf(e1 - mh) : 0.f;
      }
      const int r0 = (rbase + c0) * TSTR;
#pragma unroll
      for (int cg = 0; cg < 16; cg++) {
        v2f Bw;                                          // B[K=code][N=dim]
        Bw.x = tile[r0        + cg * 16 + n];
        Bw.y = tile[r0 + TSTR + cg * 16 + n];
        acc_s[cg] = wmma_f32_16x16x4(Aw, Bw, acc_s[cg]);
      }
    }
    bufsel ^= 1;
  }

  // ---- deterministic intra-block combine (fixed wave order, no atomics) ----
  __syncthreads();
  if (lane == 0) {
#pragma unroll
    for (int h = 0; h < HEADS; h++) {
      stat[wv * 2 * HEADS + h]         = m_run[h];
      stat[wv * 2 * HEADS + HEADS + h] = z_run[h];
    }
  }
  __syncthreads();
  float mb[HEADS], zb[HEADS];
#pragma unroll
  for (int h = 0; h < HEADS; h++) {
    float m = -3.0e38f;
    for (int w8 = 0; w8 < NWAVES; w8++) m = fmaxf(m, stat[w8 * 2 * HEADS + h]);
    mb[h] = m;
    float z = 0.f;
    for (int w8 = 0; w8 < NWAVES; w8++)
      z += stat[w8 * 2 * HEADS + HEADS + h] * __expf(stat[w8 * 2 * HEADS + h] - m);
    zb[h] = z;
  }
  float wscl[HEADS];
#pragma unroll
  for (int h = 0; h < HEADS; h++) wscl[h] = __expf(m_run[h] - mb[h]);

  float* swv = lds + LDS_TILE0;  // reuse tile0 LDS: NWAVES x 1024 scaled partial s
  if (half == 0) {               // lanes 0..15 hold rows M=0..7 (heads in elems 0..3)
#pragma unroll
    for (int cg = 0; cg < 16; cg++) {
      swv[wv * 1024 + 0 * CODE_DIM + cg * 16 + n] = acc_s[cg][0] * wscl[0];
      swv[wv * 1024 + 1 * CODE_DIM + cg * 16 + n] = acc_s[cg][1] * wscl[1];
      swv[wv * 1024 + 2 * CODE_DIM + cg * 16 + n] = acc_s[cg][2] * wscl[2];
      swv[wv * 1024 + 3 * CODE_DIM + cg * 16 + n] = acc_s[cg][3] * wscl[3];
    }
  }
  __syncthreads();
  float* rec = recs + (size_t)blockIdx.x * REC_STRIDE;
  if (t == 0) {
#pragma unroll
    for (int h = 0; h < HEADS; h++) { rec[h] = mb[h]; rec[HEADS + h] = zb[h]; }
  }
  for (int idx = t; idx < HEADS * CODE_DIM; idx += THREADS) {
    float s = 0.f;
#pragma unroll
    for (int w8 = 0; w8 < NWAVES; w8++) s += swv[w8 * 1024 + idx];
    rec[2 * HEADS + idx] = s;
  }
}

// ---------------------------------------------------------------------------
// Kernel 3: global combine + agg@W + logits.
// out[0:256) = specialty_logits, out[256:768) = multi_head_output
// ---------------------------------------------------------------------------
__global__ __launch_bounds__(THREADS) void gat_finalize_kernel(
    const float* __restrict__ recs,
    const float* __restrict__ W,  const float* __restrict__ b,
    const float* __restrict__ Wc, const float* __restrict__ bc,
    float* __restrict__ out)
{
  __shared__ float Ms[HEADS], Zs[HEADS];
  __shared__ float scl[NBLOCKS * HEADS];
  __shared__ float s_fin[HEADS * CODE_DIM];
  __shared__ float mho[HEADS * HIDDEN];
  const int t = threadIdx.x;

  if (t < HEADS) {
    float m = -3.0e38f;
    for (int r = 0; r < NBLOCKS; r++)
      m = fmaxf(m, recs[(size_t)r * REC_STRIDE + t]);
    Ms[t] = m;
  }
  __syncthreads();
  for (int idx = t; idx < NBLOCKS * HEADS; idx += THREADS) {
    const int r = idx / HEADS, h = idx % HEADS;
    scl[r * HEADS + h] = __expf(recs[(size_t)r * REC_STRIDE + h] - Ms[h]);
  }
  __syncthreads();
  if (t < HEADS) {
    float z = 0.f;
    for (int r = 0; r < NBLOCKS; r++)
      z += recs[(size_t)r * REC_STRIDE + HEADS + t] * scl[r * HEADS + t];
    Zs[t] = z;
  }
  __syncthreads();
  for (int idx = t; idx < HEADS * CODE_DIM; idx += THREADS) {
    const int h = idx >> 8;
    float acc = 0.f;
    for (int r = 0; r < NBLOCKS; r++)
      acc += recs[(size_t)r * REC_STRIDE + 2 * HEADS + idx] * scl[r * HEADS + h];
    s_fin[idx] = acc / Zs[h];                       // alpha-weighted mean code emb
  }
  __syncthreads();
  for (int idx = t; idx < HEADS * HIDDEN; idx += THREADS) {
    const int h = idx >> 7, k = idx & (HIDDEN - 1);
    float acc = b[idx];
    const float* sh = s_fin + h * CODE_DIM;
    const float* wp = W + (size_t)h * CODE_DIM * HIDDEN + k;
    for (int d = 0; d < CODE_DIM; d++) acc += sh[d] * wp[(size_t)d * HIDDEN];
    mho[idx] = acc;
    out[N_SPEC + idx] = acc;                        // multi_head_output
  }
  __syncthreads();
  for (int j = t; j < N_SPEC; j += THREADS) {
    float acc = bc[j];
    const float* wr = Wc + (size_t)j * (HEADS * HIDDEN);
    for (int q = 0; q < HEADS * HIDDEN; q++) acc += mho[q] * wr[q];
    out[j] = acc;                                   // specialty_logits
  }
}

// ---------------------------------------------------------------------------
extern "C" void kernel_launch(void* const* d_in, const int* in_sizes, int n_in,
                              void* d_out, int out_size, void* d_ws, size_t ws_size,
                              hipStream_t stream) {
  const float* pe    = (const float*)d_in[0];
  const float* codes = (const float*)d_in[1];
  const float* W     = (const float*)d_in[2];
  const float* b     = (const float*)d_in[3];
  const float* a     = (const float*)d_in[4];
  const float* Wc    = (const float*)d_in[5];
  const float* bc    = (const float*)d_in[6];
  float* ws   = (float*)d_ws;
  float* recs = ws + 2048;                 // records at +8KB; total ws use ~1.6MB

  gat_prep_kernel<<<1, THREADS, 0, stream>>>(pe, W, b, a, ws);
  const size_t ldsBytes = (size_t)LDS_TOTAL * sizeof(float);  // ~273 KB of 320 KB WGP LDS
  gat_stream_kernel<<<NBLOCKS, THREADS, ldsBytes, stream>>>(codes, ws, recs);
  gat_finalize_kernel<<<1, THREADS, 0, stream>>>(recs, W, b, Wc, bc, (float*)d_out);

  (void)in_sizes; (void)n_in; (void)out_size; (void)ws_size;
}